// TopoGAT_27023934227032
// MI455X (gfx1250) — compile-verified
//
#include <hip/hip_runtime.h>
#include <math.h>

#define FIN      128
#define TOPOF    8
#define FCAT     136   // FIN + TOPO, multiple of 4
#define H1DIM    64    // 8 heads * 8 ch
#define NHEAD    8
#define HID      8
#define COUT     40
#define COUTP    48    // padded to 3 tiles of 16
#define NEG_SLOPE 0.2f

typedef float v2f __attribute__((ext_vector_type(2)));
typedef float v8f __attribute__((ext_vector_type(8)));

// ---------- helpers ----------
__device__ __forceinline__ unsigned f32_ord_enc(float f) {
  unsigned u = __float_as_uint(f);
  return (u & 0x80000000u) ? ~u : (u | 0x80000000u);
}
__device__ __forceinline__ float f32_ord_dec(unsigned u) {
  unsigned b = (u & 0x80000000u) ? (u & 0x7FFFFFFFu) : ~u;
  return __uint_as_float(b);
}
__device__ __forceinline__ void edge_sd(const int* ei, int t, int E, int& s, int& d) {
  if (t < E) { s = ei[t]; d = ei[E + t]; }
  else       { s = t - E; d = t - E; }      // self-loops appended
}

// ---------- init ----------
__global__ void fill_f32(float* p, float v, int n) {
  int i = blockIdx.x * blockDim.x + threadIdx.x;
  if (i < n) p[i] = v;
}
__global__ void fill_u32(unsigned* p, unsigned v, int n) {
  int i = blockIdx.x * blockDim.x + threadIdx.x;
  if (i < n) p[i] = v;
}

// ---------- prep ----------
__global__ void concat_kernel(const float* __restrict__ x, const float* __restrict__ topo,
                              float* __restrict__ hcat, int N) {
  int i = blockIdx.x * blockDim.x + threadIdx.x;
  if (i >= N * FCAT) return;
  int n = i / FCAT, j = i - n * FCAT;
  hcat[i] = (j < FIN) ? x[n * FIN + j] : topo[n * TOPOF + (j - FIN)];
}
__global__ void transpose_w1(const float* __restrict__ W, float* __restrict__ Wt) {
  int i = blockIdx.x * blockDim.x + threadIdx.x;      // [FCAT,H1DIM] -> [H1DIM,FCAT]
  if (i >= FCAT * H1DIM) return;
  int k = i / H1DIM, n = i - k * H1DIM;
  Wt[n * FCAT + k] = W[k * H1DIM + n];
}
__global__ void transpose_w2(const float* __restrict__ W, float* __restrict__ Wt) {
  int i = blockIdx.x * blockDim.x + threadIdx.x;      // [H1DIM,COUT] -> [COUTP,H1DIM], zero pad
  if (i >= COUTP * H1DIM) return;
  int n = i / H1DIM, k = i - n * H1DIM;
  Wt[n * H1DIM + k] = (n < COUT) ? W[k * COUT + n] : 0.0f;
}

// ---------- WMMA GEMM: C[M,Nout] = A[M,K] * Bt[N,K]^T ----------
// One wave owns a 16-row block of A and ALL TN column tiles: the A fragment is
// loaded once per K-step and shared by TN consecutive WMMAs (A traffic 1x).
template <int K, int TN>
__global__ void wmma_gemm_f32(const float* __restrict__ A, const float* __restrict__ Bt,
                              float* __restrict__ C, int Nout, int total_mtiles) {
  int wave = (blockIdx.x * blockDim.x + threadIdx.x) >> 5;
  if (wave >= total_mtiles) return;                   // wave-uniform exit
  int lane = threadIdx.x & 31;
  int half = lane >> 4, r = lane & 15;
  const float* arow = A + (size_t)(wave * 16 + r) * K;
  const float* brow[TN];
  v8f acc[TN];
#pragma unroll
  for (int t = 0; t < TN; ++t) {
    brow[t] = Bt + (size_t)(t * 16 + r) * K;
    acc[t] = (v8f){};
  }
#pragma unroll
  for (int k0 = 0; k0 < K; k0 += 4) {
    int kk = k0 + 2 * half;
    v2f a = *(const v2f*)(arow + kk);                 // A: M=r, K=2*half+{0,1}
#pragma unroll
    for (int t = 0; t < TN; ++t) {
      v2f b = *(const v2f*)(brow[t] + kk);            // B: N=r, K=2*half+{0,1}
      acc[t] = __builtin_amdgcn_wmma_f32_16x16x4_f32(false, a, false, b, (short)0,
                                                     acc[t], false, false);
    }
  }
#pragma unroll
  for (int t = 0; t < TN; ++t) {
    int col = t * 16 + r;
    if (col < Nout) {
#pragma unroll
      for (int i = 0; i < 8; ++i) {                   // D: VGPR i -> M = i + 8*half
        int row = wave * 16 + i + 8 * half;
        C[(size_t)row * Nout + col] = acc[t][i];
      }
    }
  }
}

// ---------- attention coefficients ----------
__global__ void alpha1_kernel(const float* __restrict__ H, const float* __restrict__ a_s,
                              const float* __restrict__ a_d, float* __restrict__ as,
                              float* __restrict__ ad, int N) {
  int t = blockIdx.x * blockDim.x + threadIdx.x;      // (node, head)
  if (t >= N * NHEAD) return;
  int n = t >> 3, h = t & 7;
  const float* hp = H + (size_t)n * H1DIM + h * HID;
  float s = 0.f, d = 0.f;
#pragma unroll
  for (int c = 0; c < HID; ++c) { s += hp[c] * a_s[h * HID + c]; d += hp[c] * a_d[h * HID + c]; }
  as[t] = s; ad[t] = d;
}
__global__ void alpha2_kernel(const float* __restrict__ H, const float* __restrict__ a_s,
                              const float* __restrict__ a_d, float* __restrict__ as,
                              float* __restrict__ ad, int N) {
  int n = blockIdx.x * blockDim.x + threadIdx.x;
  if (n >= N) return;
  const float* hp = H + (size_t)n * COUT;
  float s = 0.f, d = 0.f;
#pragma unroll
  for (int c = 0; c < COUT; ++c) { s += hp[c] * a_s[c]; d += hp[c] * a_d[c]; }
  as[n] = s; ad[n] = d;
}

// ---------- layer-1 edge passes (8 heads) ----------
__global__ void edge_max1(const int* __restrict__ ei, const float* __restrict__ as,
                          const float* __restrict__ ad, float* __restrict__ ebuf,
                          unsigned* __restrict__ m, int E, int Etot) {
  int t = blockIdx.x * blockDim.x + threadIdx.x;
  if (t >= Etot * NHEAD) return;
  int e = t >> 3, h = t & 7, s, d;
  edge_sd(ei, e, E, s, d);
  float v = as[s * NHEAD + h] + ad[d * NHEAD + h];
  v = (v > 0.f) ? v : NEG_SLOPE * v;
  ebuf[t] = v;
  atomicMax(&m[d * NHEAD + h], f32_ord_enc(v));
}
__global__ void edge_exp1(const int* __restrict__ ei, float* __restrict__ ebuf,
                          const unsigned* __restrict__ m, float* __restrict__ z,
                          int E, int Etot) {
  int t = blockIdx.x * blockDim.x + threadIdx.x;
  if (t >= Etot * NHEAD) return;
  int e = t >> 3, h = t & 7, s, d;
  edge_sd(ei, e, E, s, d);
  float ex = __expf(ebuf[t] - f32_ord_dec(m[d * NHEAD + h]));
  ebuf[t] = ex;
  atomicAdd(&z[d * NHEAD + h], ex);
}
__global__ void edge_aggr1(const int* __restrict__ ei, const float* __restrict__ ebuf,
                           const float* __restrict__ z, const float* __restrict__ H,
                           float* __restrict__ aggr, int E, int Etot) {
  int t = blockIdx.x * blockDim.x + threadIdx.x;
  if (t >= Etot * NHEAD) return;
  int e = t >> 3, h = t & 7, s, d;
  edge_sd(ei, e, E, s, d);
  float alpha = ebuf[t] / (z[d * NHEAD + h] + 1e-16f);
  const float* hp = H + (size_t)s * H1DIM + h * HID;
  float* op = aggr + (size_t)d * H1DIM + h * HID;
#pragma unroll
  for (int c = 0; c < HID; ++c) atomicAdd(&op[c], hp[c] * alpha);
}
__global__ void elu_bias(float* __restrict__ a, const float* __restrict__ b, int N) {
  int i = blockIdx.x * blockDim.x + threadIdx.x;
  if (i >= N * H1DIM) return;
  float v = a[i] + b[i & (H1DIM - 1)];
  a[i] = (v > 0.f) ? v : (__expf(v) - 1.0f);
}

// ---------- layer-2 edge passes (1 head) ----------
__global__ void edge_max2(const int* __restrict__ ei, const float* __restrict__ as,
                          const float* __restrict__ ad, float* __restrict__ ebuf,
                          unsigned* __restrict__ m, int E, int Etot) {
  int t = blockIdx.x * blockDim.x + threadIdx.x;
  if (t >= Etot) return;
  int s, d; edge_sd(ei, t, E, s, d);
  float v = as[s] + ad[d];
  v = (v > 0.f) ? v : NEG_SLOPE * v;
  ebuf[t] = v;
  atomicMax(&m[d], f32_ord_enc(v));
}
__global__ void edge_exp2(const int* __restrict__ ei, float* __restrict__ ebuf,
                          const unsigned* __restrict__ m, float* __restrict__ z,
                          int E, int Etot) {
  int t = blockIdx.x * blockDim.x + threadIdx.x;
  if (t >= Etot) return;
  int s, d; edge_sd(ei, t, E, s, d);
  float ex = __expf(ebuf[t] - f32_ord_dec(m[d]));
  ebuf[t] = ex;
  atomicAdd(&z[d], ex);
}
__global__ void edge_aggr2(const int* __restrict__ ei, const float* __restrict__ ebuf,
                           const float* __restrict__ z, const float* __restrict__ H,
                           float* __restrict__ aggr, int E, int Etot) {
  int t = blockIdx.x * blockDim.x + threadIdx.x;
  if (t >= Etot) return;
  int s, d; edge_sd(ei, t, E, s, d);
  float alpha = ebuf[t] / (z[d] + 1e-16f);
  const float* hp = H + (size_t)s * COUT;
  float* op = aggr + (size_t)d * COUT;
#pragma unroll
  for (int c = 0; c < COUT; ++c) atomicAdd(&op[c], hp[c] * alpha);
}

// ---------- final log_softmax ----------
__global__ void logsoftmax_kernel(const float* __restrict__ a, const float* __restrict__ b2,
                                  float* __restrict__ out, int N) {
  int n = blockIdx.x * blockDim.x + threadIdx.x;
  if (n >= N) return;
  float v[COUT];
  float mx = -3.402823466e38f;
#pragma unroll
  for (int c = 0; c < COUT; ++c) { v[c] = a[(size_t)n * COUT + c] + b2[c]; mx = fmaxf(mx, v[c]); }
  float s = 0.f;
#pragma unroll
  for (int c = 0; c < COUT; ++c) s += __expf(v[c] - mx);
  float lse = mx + __logf(s);
#pragma unroll
  for (int c = 0; c < COUT; ++c) out[(size_t)n * COUT + c] = v[c] - lse;
}

static inline int ceil_div(long a, long b) { return (int)((a + b - 1) / b); }

extern "C" void kernel_launch(void* const* d_in, const int* in_sizes, int n_in,
                              void* d_out, int out_size, void* d_ws, size_t ws_size,
                              hipStream_t stream) {
  const float* x      = (const float*)d_in[0];
  const float* topo   = (const float*)d_in[1];
  const int*   ei     = (const int*)  d_in[2];
  const float* W1     = (const float*)d_in[3];
  const float* a_src1 = (const float*)d_in[4];
  const float* a_dst1 = (const float*)d_in[5];
  const float* b1     = (const float*)d_in[6];
  const float* W2     = (const float*)d_in[7];
  const float* a_src2 = (const float*)d_in[8];
  const float* a_dst2 = (const float*)d_in[9];
  const float* b2     = (const float*)d_in[10];
  float* out = (float*)d_out;

  const int N = in_sizes[0] / FIN;      // 100000 (multiple of 16)
  const int E = in_sizes[2] / 2;        // 1600000
  const int Etot = E + N;

  // workspace arena (hcat region reused for layer-1 edge buffer after GEMM1)
  char* base = (char*)d_ws;
  size_t off = 0;
  auto arena = [&](size_t bytes) -> char* {
    char* p = base + off;
    off = (off + bytes + 255) & ~(size_t)255;
    return p;
  };
  size_t hcat_b = (size_t)N * FCAT * 4, ebuf1_b = (size_t)Etot * NHEAD * 4;
  char*  shared0 = arena(hcat_b > ebuf1_b ? hcat_b : ebuf1_b);
  float* hcat  = (float*)shared0;       // live until GEMM1 done
  float* ebuf1 = (float*)shared0;       // live after GEMM1
  float* W1t   = (float*)arena((size_t)H1DIM * FCAT * 4);
  float* W2t   = (float*)arena((size_t)COUTP * H1DIM * 4);
  float* H1    = (float*)arena((size_t)N * H1DIM * 4);
  float* as1   = (float*)arena((size_t)N * NHEAD * 4);
  float* ad1   = (float*)arena((size_t)N * NHEAD * 4);
  unsigned* m1 = (unsigned*)arena((size_t)N * NHEAD * 4);
  float* z1    = (float*)arena((size_t)N * NHEAD * 4);
  float* aggr1 = (float*)arena((size_t)N * H1DIM * 4);
  float* H2    = (float*)arena((size_t)N * COUT * 4);
  float* as2   = (float*)arena((size_t)N * 4);
  float* ad2   = (float*)arena((size_t)N * 4);
  unsigned* m2 = (unsigned*)arena((size_t)N * 4);
  float* z2    = (float*)arena((size_t)N * 4);
  float* ebuf2 = (float*)arena((size_t)Etot * 4);
  float* aggr2 = (float*)arena((size_t)N * COUT * 4);

  const int T = 256;
  // prep
  transpose_w1<<<ceil_div((long)FCAT * H1DIM, T), T, 0, stream>>>(W1, W1t);
  transpose_w2<<<ceil_div((long)COUTP * H1DIM, T), T, 0, stream>>>(W2, W2t);
  concat_kernel<<<ceil_div((long)N * FCAT, T), T, 0, stream>>>(x, topo, hcat, N);
  fill_u32<<<ceil_div((long)N * NHEAD, T), T, 0, stream>>>(m1, 0u, N * NHEAD);
  fill_f32<<<ceil_div((long)N * NHEAD, T), T, 0, stream>>>(z1, 0.f, N * NHEAD);
  fill_f32<<<ceil_div((long)N * H1DIM, T), T, 0, stream>>>(aggr1, 0.f, N * H1DIM);
  fill_u32<<<ceil_div((long)N, T), T, 0, stream>>>(m2, 0u, N);
  fill_f32<<<ceil_div((long)N, T), T, 0, stream>>>(z2, 0.f, N);
  fill_f32<<<ceil_div((long)N * COUT, T), T, 0, stream>>>(aggr2, 0.f, N * COUT);

  // layer 1: GEMM (WMMA f32, A loaded once, 4 column tiles per wave), attention, ELU
  int mtiles = N / 16;
  wmma_gemm_f32<FCAT, H1DIM / 16><<<ceil_div((long)mtiles, T / 32), T, 0, stream>>>(
      hcat, W1t, H1, H1DIM, mtiles);
  alpha1_kernel<<<ceil_div((long)N * NHEAD, T), T, 0, stream>>>(H1, a_src1, a_dst1, as1, ad1, N);
  edge_max1 <<<ceil_div((long)Etot * NHEAD, T), T, 0, stream>>>(ei, as1, ad1, ebuf1, m1, E, Etot);
  edge_exp1 <<<ceil_div((long)Etot * NHEAD, T), T, 0, stream>>>(ei, ebuf1, m1, z1, E, Etot);
  edge_aggr1<<<ceil_div((long)Etot * NHEAD, T), T, 0, stream>>>(ei, ebuf1, z1, H1, aggr1, E, Etot);
  elu_bias  <<<ceil_div((long)N * H1DIM, T), T, 0, stream>>>(aggr1, b1, N);

  // layer 2: GEMM (WMMA f32, padded 48 cols, stores guarded to 40), attention, log_softmax
  wmma_gemm_f32<H1DIM, COUTP / 16><<<ceil_div((long)mtiles, T / 32), T, 0, stream>>>(
      aggr1, W2t, H2, COUT, mtiles);
  alpha2_kernel<<<ceil_div((long)N, T), T, 0, stream>>>(H2, a_src2, a_dst2, as2, ad2, N);
  edge_max2 <<<ceil_div((long)Etot, T), T, 0, stream>>>(ei, as2, ad2, ebuf2, m2, E, Etot);
  edge_exp2 <<<ceil_div((long)Etot, T), T, 0, stream>>>(ei, ebuf2, m2, z2, E, Etot);
  edge_aggr2<<<ceil_div((long)Etot, T), T, 0, stream>>>(ei, ebuf2, z2, H2, aggr2, E, Etot);
  logsoftmax_kernel<<<ceil_div((long)N, T), T, 0, stream>>>(aggr2, b2, out, N);
}